// SequenceGroupAggregating_59055800320846
// MI455X (gfx1250) — compile-verified
//
#include <hip/hip_runtime.h>
#include <hip/hip_bf16.h>

typedef __attribute__((ext_vector_type(2))) float v2f;
typedef __attribute__((ext_vector_type(8))) float v8f;

#define B_ 16
#define S_ 2048
#define H_ 1024
#define A_ 1024

// ---------------------------------------------------------------------------
// Kernel 1: segment boundaries. lb[b][a] = first s with group_by[b][s] >= a,
// for a in [0, A]. group_by is sorted per batch, so group a occupies
// s in [lb[a], lb[a+1]).
// ---------------------------------------------------------------------------
__global__ void seg_bounds_kernel(const int* __restrict__ gb, int* __restrict__ lb) {
    const int per = A_ + 1;
    int idx = blockIdx.x * blockDim.x + threadIdx.x;
    if (idx >= B_ * per) return;
    int b = idx / per;
    int a = idx - b * per;
    const int* g = gb + b * S_;
    int lo = 0, hi = S_;
    while (lo < hi) {
        int mid = (lo + hi) >> 1;
        if (g[mid] < a) lo = mid + 1; else hi = mid;
    }
    lb[idx] = lo;
}

// ---------------------------------------------------------------------------
// Kernel 2: banded segment-mean via V_WMMA_F32_16X16X4_F32.
// Block = 8 waves, one (batch, 16-group tile). Wave owns 2x 64-column quads.
// Per 4-source-row step: A = 16x4 one-hot from group_by compares (exact),
// B = 4x16 slice of x, D += A*B in full f32. Scale by 1/count at the end.
// ---------------------------------------------------------------------------
__global__ void __launch_bounds__(256)
seg_agg_wmma_kernel(const float* __restrict__ x, const int* __restrict__ gb,
                    const int* __restrict__ lb, float* __restrict__ out) {
    const int bt   = blockIdx.x;        // 0 .. B*(A/16)-1
    const int b    = bt >> 6;           // A/16 == 64 tiles per batch
    const int a0   = (bt & 63) << 4;
    const int lane = threadIdx.x & 31;
    const int wave = threadIdx.x >> 5;
    const int half = lane >> 4;         // 0: K=0,1 / M,N lanes 0-15 ; 1: K=2,3
    const int n    = lane & 15;
    const int k0   = half << 1;

    const int* lbb = lb + b * (A_ + 1);
    const int segStart = lbb[a0];
    const int segEnd   = lbb[a0 + 16];

    // Inverse counts for the output rows this lane holds: m = r + 8*half.
    float invc[8];
#pragma unroll
    for (int r = 0; r < 8; ++r) {
        int a = a0 + (half << 3) + r;
        int cnt = lbb[a + 1] - lbb[a];
        invc[r] = (cnt > 0) ? (1.0f / (float)cnt) : 0.0f;
    }

    const int*   gbb = gb + b * S_;
    const float* xb  = x + (size_t)b * S_ * H_;
    float*       ob  = out + ((size_t)b * A_ + a0) * H_;

    const int aRow = a0 + n;            // one-hot compare target (A row = lane N)

    for (int hq = wave; hq < 16; hq += 8) {
        const int h0 = hq << 6;         // 64 columns per wave pass
        v8f acc0 = {}, acc1 = {}, acc2 = {}, acc3 = {};

        for (int s = segStart; s < segEnd; s += 4) {
            int  s0 = s + k0;
            int  s1 = s0 + 1;
            bool v0 = s0 < segEnd;
            bool v1 = s1 < segEnd;
            int  s0c = v0 ? s0 : (segEnd - 1);
            int  s1c = v1 ? s1 : (segEnd - 1);

            int g0 = gbb[s0c];
            int g1 = gbb[s1c];
            v2f amat;
            amat.x = (v0 && g0 == aRow) ? 1.0f : 0.0f;
            amat.y = (v1 && g1 == aRow) ? 1.0f : 0.0f;

            const float* r0 = xb + (size_t)s0c * H_ + h0 + n;
            const float* r1 = xb + (size_t)s1c * H_ + h0 + n;

            v2f bm;
            bm.x = r0[0];  bm.y = r1[0];
            acc0 = __builtin_amdgcn_wmma_f32_16x16x4_f32(false, amat, false, bm,
                                                         (short)0, acc0, false, false);
            bm.x = r0[16]; bm.y = r1[16];
            acc1 = __builtin_amdgcn_wmma_f32_16x16x4_f32(false, amat, false, bm,
                                                         (short)0, acc1, false, false);
            bm.x = r0[32]; bm.y = r1[32];
            acc2 = __builtin_amdgcn_wmma_f32_16x16x4_f32(false, amat, false, bm,
                                                         (short)0, acc2, false, false);
            bm.x = r0[48]; bm.y = r1[48];
            acc3 = __builtin_amdgcn_wmma_f32_16x16x4_f32(false, amat, false, bm,
                                                         (short)0, acc3, false, false);
        }

        // D layout: lane holds N=n; VGPR r holds M = r + 8*half.
#pragma unroll
        for (int r = 0; r < 8; ++r) {
            int m = r + (half << 3);
            float* orow = ob + (size_t)m * H_ + h0 + n;
            float iv = invc[r];
            orow[0]  = acc0[r] * iv;
            orow[16] = acc1[r] * iv;
            orow[32] = acc2[r] * iv;
            orow[48] = acc3[r] * iv;
        }
    }
}

extern "C" void kernel_launch(void* const* d_in, const int* in_sizes, int n_in,
                              void* d_out, int out_size, void* d_ws, size_t ws_size,
                              hipStream_t stream) {
    const float* x  = (const float*)d_in[0];
    const int*   gb = (const int*)d_in[1];
    // d_in[2] is agg_step == 1024, baked in as A_.
    (void)in_sizes; (void)n_in; (void)out_size; (void)ws_size;

    int*   lb  = (int*)d_ws;            // B*(A+1) ints = 65,600 bytes of scratch
    float* out = (float*)d_out;

    const int nBound = B_ * (A_ + 1);
    seg_bounds_kernel<<<(nBound + 255) / 256, 256, 0, stream>>>(gb, lb);
    seg_agg_wmma_kernel<<<B_ * (A_ / 16), 256, 0, stream>>>(x, gb, lb, out);
}